// EnhancedMaskLoss_90494960927125
// MI455X (gfx1250) — compile-verified
//
#include <hip/hip_runtime.h>
#include <math.h>

// Problem constants (match reference)
#define B_    4
#define Q_    100
#define M_    30
#define P_    80000
#define NP_   45000
#define NC_   20          // NUM_CLASSES; no-object class id == NC_
#define EOS_COEF_ 0.1f
#define W_CE_   2.0f
#define W_DICE_ 5.0f
#define W_MASK_ 5.0f

#define NWORDS ((P_ + 31) / 32)   // 2500 32-bit words (P_ % 32 == 0, NWORDS % 4 == 0)

#define CE_THREADS 512
#define CE_WAVES   (CE_THREADS / 32)

#define MD_THREADS 1024
#define MD_WAVES   (MD_THREADS / 32)
#define GSTRIDE    (MD_WAVES * 4)   // words consumed per block per pipeline step

__device__ __forceinline__ float wave_reduce_add(float v) {
#pragma unroll
  for (int off = 16; off > 0; off >>= 1) v += __shfl_xor(v, off, 32);
  return v;
}

// ---------------------------------------------------------------------------
// Kernel 1: weighted cross-entropy over [B,Q,NC+1] logits. One block.
// ws[0] = sum(w * nll), ws[1] = sum(w)
// ---------------------------------------------------------------------------
__global__ __launch_bounds__(CE_THREADS)
void ce_kernel(const float* __restrict__ logits,
               const int* __restrict__ tclasses,
               const int* __restrict__ src_idx,
               const int* __restrict__ tgt_idx,
               float* __restrict__ ws) {
  __shared__ int s_src[B_ * M_];
  __shared__ int s_cls_of_src[B_ * M_];   // gathered class for each match slot
  __shared__ float redA[CE_WAVES];
  __shared__ float redB[CE_WAVES];
  const int tid = threadIdx.x;

  if (tid < B_ * M_) {
    const int b = tid / M_;
    s_src[tid] = src_idx[tid];
    s_cls_of_src[tid] = tclasses[b * M_ + tgt_idx[tid]];
  }
  __syncthreads();

  float wn = 0.0f, wsum = 0.0f;
  if (tid < B_ * Q_) {
    const int b = tid / Q_;
    const int q = tid % Q_;
    int cls = NC_;                         // default: no-object
#pragma unroll
    for (int mi = 0; mi < M_; ++mi) {
      if (s_src[b * M_ + mi] == q) cls = s_cls_of_src[b * M_ + mi];
    }
    const float* lg = logits + (size_t)(b * Q_ + q) * (NC_ + 1);
    float mx = lg[0];
#pragma unroll
    for (int c1 = 1; c1 <= NC_; ++c1) mx = fmaxf(mx, lg[c1]);
    float se = 0.0f;
#pragma unroll
    for (int c1 = 0; c1 <= NC_; ++c1) se += expf(lg[c1] - mx);
    const float lse = mx + logf(se);
    const float nll = lse - lg[cls];
    const float w = (cls == 0) ? 0.0f : ((cls == NC_) ? EOS_COEF_ : 1.0f);
    wn = w * nll;
    wsum = w;
  }
  wn = wave_reduce_add(wn);
  wsum = wave_reduce_add(wsum);
  const int wave = tid >> 5, lane = tid & 31;
  if (lane == 0) { redA[wave] = wn; redB[wave] = wsum; }
  __syncthreads();
  if (tid == 0) {
    float a = 0.0f, bsum = 0.0f;
#pragma unroll
    for (int i = 0; i < CE_WAVES; ++i) { a += redA[i]; bsum += redB[i]; }
    ws[0] = a;
    ws[1] = bsum;
  }
}

// ---------------------------------------------------------------------------
// Kernel 2: per matched mask (b,m): BCE sum + dice partial sums.
// Phase A: stream the binary 320KB target row through LDS with software-
//          pipelined CDNA5 async global->LDS copies (double buffered; async
//          loads complete in order, so s_wait_asynccnt 4 retires the older
//          group), ballot-pack (wave32) into a 10KB LDS bitset.
// Phase B: NP point samples: scattered pred gather + 1-bit LDS lookup.
// Output: ws[2 + bm*4 + {0..3}] = {sum_bce, sum_p, sum_p_st, sum_st}
// ---------------------------------------------------------------------------
__global__ __launch_bounds__(MD_THREADS)
void maskdice_kernel(const float* __restrict__ pred_masks,
                     const float* __restrict__ target_masks,
                     const int* __restrict__ src_idx,
                     const int* __restrict__ tgt_idx,
                     const int* __restrict__ sample_idx,
                     float* __restrict__ ws) {
  __shared__ unsigned bits[NWORDS];              // 10000 B bitset
  __shared__ float stage[MD_WAVES * 8 * 32];     // 32 KB: 2 groups x 4 words / wave
  __shared__ float red[MD_WAVES][4];

  const int tid = threadIdx.x;
  const int wave = tid >> 5, lane = tid & 31;
  const int bm = blockIdx.x;
  const int b = bm / M_;
  const int m = bm % M_;

  const int c = src_idx[b * M_ + m];                       // matched query col
  const int r = tgt_idx[b * M_ + m];                       // permuted GT row
  const float* row = target_masks + ((size_t)b * M_ + r) * (size_t)P_;

  // ---- Phase A: pipelined async row -> LDS, ballot-pack ----
  const unsigned wslot = (unsigned)wave * 8u * 32u;
  auto issue_group = [&](int wb, int buf) {
#pragma unroll
    for (int j = 0; j < 4; ++j) {
      unsigned ldsoff = (unsigned)(size_t)
          (&stage[wslot + ((unsigned)buf * 4u + (unsigned)j) * 32u + (unsigned)lane]);
      unsigned long long gaddr = (unsigned long long)(size_t)
          (row + ((size_t)(wb + j) * 32 + lane));
      asm volatile("global_load_async_to_lds_b32 %0, %1, off"
                   :: "v"(ldsoff), "v"(gaddr) : "memory");
    }
  };
  auto process_group = [&](int wb, int buf) {
#pragma unroll
    for (int j = 0; j < 4; ++j) {
      const float v = stage[wslot + ((unsigned)buf * 4u + (unsigned)j) * 32u + (unsigned)lane];
      const unsigned mb = __builtin_amdgcn_ballot_w32(v > 0.5f);
      if (lane == 0) bits[wb + j] = mb;
    }
  };

  int buf = 0;
  int wb = wave * 4;
  if (wb < NWORDS) issue_group(wb, buf);
#pragma unroll 1
  for (; wb < NWORDS; wb += GSTRIDE) {
    const int nxt = wb + GSTRIDE;
    if (nxt < NWORDS) {
      issue_group(nxt, buf ^ 1);
      asm volatile("s_wait_asynccnt 0x4" ::: "memory");   // older group retired
    } else {
      asm volatile("s_wait_asynccnt 0x0" ::: "memory");
    }
    process_group(wb, buf);
    buf ^= 1;
  }
  __syncthreads();

  // ---- Phase B: point sampling + elementwise BCE / dice partials ----
  const int* samp = sample_idx + (size_t)b * NP_;
  const float* predcol = pred_masks + (size_t)b * P_ * Q_ + c;

  float s_bce = 0.0f, s_p = 0.0f, s_pst = 0.0f, s_st = 0.0f;
#pragma unroll 4
  for (int n = tid; n < NP_; n += MD_THREADS) {
    const int idx = samp[n];
    const float sp = predcol[(size_t)idx * Q_];
    const unsigned bit = (bits[idx >> 5] >> (idx & 31)) & 1u;

    const float e = expf(-fabsf(sp));          // shared by log1p & sigmoid
    const float l1 = log1pf(e);
    const float inv = 1.0f / (1.0f + e);
    float p = (sp >= 0.0f) ? inv : e * inv;    // sigmoid(sp)
    p = fminf(fmaxf(p, 1e-7f), 1.0f - 1e-7f);

    const float st = (float)bit;
    s_bce += fmaxf(sp, 0.0f) - sp * st + l1;
    s_p += p;
    if (bit) { s_pst += p; s_st += 1.0f; }
  }

  s_bce = wave_reduce_add(s_bce);
  s_p   = wave_reduce_add(s_p);
  s_pst = wave_reduce_add(s_pst);
  s_st  = wave_reduce_add(s_st);
  if (lane == 0) {
    red[wave][0] = s_bce; red[wave][1] = s_p;
    red[wave][2] = s_pst; red[wave][3] = s_st;
  }
  __syncthreads();
  if (tid == 0) {
    float a0 = 0.0f, a1 = 0.0f, a2 = 0.0f, a3 = 0.0f;
#pragma unroll
    for (int i = 0; i < MD_WAVES; ++i) {
      a0 += red[i][0]; a1 += red[i][1]; a2 += red[i][2]; a3 += red[i][3];
    }
    float* o = ws + 2 + (size_t)bm * 4;
    o[0] = a0; o[1] = a1; o[2] = a2; o[3] = a3;
  }
}

// ---------------------------------------------------------------------------
// Kernel 3: deterministic fixed-order final reduction -> d_out[3]
// ---------------------------------------------------------------------------
__global__ void finalize_kernel(const float* __restrict__ ws,
                                float* __restrict__ out) {
  const float loss_ce = ws[0] / fmaxf(ws[1], 1e-8f);
  float sm = 0.0f, sd = 0.0f;
#pragma unroll 1
  for (int bm = 0; bm < B_ * M_; ++bm) {
    const float* s = ws + 2 + (size_t)bm * 4;
    sm += s[0] / (float)NP_;                                   // bce mean
    sd += 1.0f - (2.0f * s[2] + 1.0f) / (s[1] + s[3] + 1.0f);  // dice term
  }
  const float inv_nm = 1.0f / (float)(B_ * M_);
  out[0] = W_CE_ * loss_ce;
  out[1] = W_DICE_ * (sd * inv_nm);
  out[2] = W_MASK_ * (sm * inv_nm);
}

extern "C" void kernel_launch(void* const* d_in, const int* in_sizes, int n_in,
                              void* d_out, int out_size, void* d_ws, size_t ws_size,
                              hipStream_t stream) {
  (void)in_sizes; (void)n_in; (void)out_size; (void)ws_size;
  const float* pred_logits    = (const float*)d_in[0];
  const float* pred_masks     = (const float*)d_in[1];
  const float* target_masks   = (const float*)d_in[2];
  const int*   target_classes = (const int*)d_in[3];
  const int*   src_idx        = (const int*)d_in[4];
  const int*   tgt_idx        = (const int*)d_in[5];
  const int*   sample_idx     = (const int*)d_in[6];
  float* ws  = (float*)d_ws;
  float* out = (float*)d_out;

  ce_kernel<<<1, CE_THREADS, 0, stream>>>(pred_logits, target_classes,
                                          src_idx, tgt_idx, ws);
  maskdice_kernel<<<B_ * M_, MD_THREADS, 0, stream>>>(pred_masks, target_masks,
                                                      src_idx, tgt_idx, sample_idx, ws);
  finalize_kernel<<<1, 1, 0, stream>>>(ws, out);
}